// Layer_9345848836447
// MI455X (gfx1250) — compile-verified
//
#include <hip/hip_runtime.h>
#include <math.h>

#define DDIM     128
#define N_OUT    100000
#define N_EDGES  1600000
#define NSTRIPES (N_OUT / 16)     // 6250
#define WAVES_PB 8
#define WPITCH   288              // pair-row pitch (floats): 256 + 32 pad
                                  // (+128B) so hi=0/hi=1 half-waves hit
                                  // disjoint LDS bank sets on b64 loads

typedef __attribute__((ext_vector_type(2))) float v2f;
typedef __attribute__((ext_vector_type(8))) float v8f;

static __device__ __forceinline__ float fast_tanh(float x) {
#if __has_builtin(__builtin_amdgcn_tanhf)
    return __builtin_amdgcn_tanhf(x);   // V_TANH_F32 hardware transcendental
#else
    return tanhf(x);
#endif
}

// ---------------------------------------------------------------------------
// Kernel 1: init stripe_start[] = N_EDGES (stripes with no edges keep this)
// ---------------------------------------------------------------------------
__global__ void init_stripe_kernel(int* __restrict__ stripe_start) {
    for (int i = threadIdx.x; i < NSTRIPES; i += blockDim.x)
        stripe_start[i] = N_EDGES;
}

// ---------------------------------------------------------------------------
// Kernel 2: boundary marking on the sorted segment_ids.
// stripe_start[j] = min e such that segid[e] >= 16*j.
// Edge e is that minimum for all j with segid[e-1] < 16*j <= segid[e].
// ---------------------------------------------------------------------------
__global__ void __launch_bounds__(256)
stripe_start_kernel(const int* __restrict__ segid,
                    int*       __restrict__ stripe_start) {
    int e = blockIdx.x * blockDim.x + threadIdx.x;
    if (e >= N_EDGES) return;
    int s_cur = segid[e];
    int j_min = (e == 0) ? 0 : ((segid[e - 1] >> 4) + 1);
    int j_max = s_cur >> 4;
    for (int j = j_min; j <= j_max; ++j) stripe_start[j] = e;
}

// ---------------------------------------------------------------------------
// Kernel 3 (fused): per wave --
//   (a) run-length gather+segment-sum of its 16 consecutive segments straight
//       into an LDS stripe (sorted ids => contiguous edge slice, no atomics),
//   (b) 16x128 @ 128x128 fp32 WMMA GEMM, A from the LDS stripe, B from a
//       K-pair-interleaved LDS copy of W: Wp[k/2][n] = {W[k][n], W[k+1][n]}
//       so every B fragment is ONE contiguous ds_load_b64 into an even-
//       aligned VGPR pair (no repacking movs), (c) hw-tanh epilogue.
// LDS: 72KB (Wp, padded) + 8 waves * 8KB (stripes) = 136KB of 320KB WGP LDS.
//
// fp32 WMMA register layouts (ISA 7.12.2):
//   A 16x4 : lane l -> M = l&15 ; VGPR0 = K[2*(l>>4)], VGPR1 = K[2*(l>>4)+1]
//   B 4x16 : lane l -> N = l&15 ; VGPR0 = K[2*(l>>4)], VGPR1 = K[2*(l>>4)+1]
//   C 16x16: lane l, VGPR r -> M = r + 8*(l>>4), N = l&15
// ---------------------------------------------------------------------------
__global__ void __launch_bounds__(256)
fused_segsum_gemm_tanh_kernel(const float* __restrict__ values,
                              const int*   __restrict__ gidx,
                              const int*   __restrict__ segid,
                              const int*   __restrict__ stripe_start,
                              const float* __restrict__ W,
                              float*       __restrict__ out) {
    __shared__ float Wlds[(DDIM / 2) * WPITCH];   // 64 pair-rows x 288
    __shared__ float Stripe[WAVES_PB][16 * DDIM];

    // cooperative W -> LDS in K-pair-interleaved layout (all 256 threads)
    for (int i = threadIdx.x; i < (DDIM / 2) * DDIM; i += blockDim.x) {
        int p = i >> 7;          // K pair index (k = 2p, 2p+1)
        int n = i & (DDIM - 1);  // column
        float2 w;
        w.x = W[(2 * p)     * DDIM + n];
        w.y = W[(2 * p + 1) * DDIM + n];
        *(float2*)&Wlds[p * WPITCH + n * 2] = w;
    }
    __syncthreads();

    const int lane   = threadIdx.x & 31;
    const int wid    = threadIdx.x >> 5;
    const int stripe = blockIdx.x * WAVES_PB + wid;
    if (stripe >= NSTRIPES) return;       // whole-wave uniform exit

    const int s0   = stripe * 16;
    const int s_hi = s0 + 16;
    const int col  = lane * 4;            // lane owns 4 consecutive columns
    float* my = &Stripe[wid][0];

    // ---- (a) zero stripe rows, then scan this stripe's contiguous edges ----
    #pragma unroll
    for (int r = 0; r < 16; ++r)
        *(float4*)(my + r * DDIM + col) = make_float4(0.f, 0.f, 0.f, 0.f);

    int e   = stripe_start[stripe];
    int cur = -1;                          // absolute segment id of open run
    float4 acc = make_float4(0.f, 0.f, 0.f, 0.f);
    bool done = (e >= N_EDGES);

    while (!done) {
        // prefetch 32 edges' metadata coalesced, broadcast lane-by-lane
        int idx = e + lane;
        int sg = 0x7FFFFFFF, gi = 0;       // sentinel terminates the scan
        if (idx < N_EDGES) { sg = segid[idx]; gi = gidx[idx]; }
        for (int j = 0; j < 32; ++j) {
            int s = __shfl(sg, j);
            if (s >= s_hi) { done = true; break; }
            int g = __shfl(gi, j);
            float4 v = *(const float4*)(values + (size_t)g * DDIM + col);
            if (s != cur) {
                if (cur >= 0)
                    *(float4*)(my + (cur - s0) * DDIM + col) = acc;
                cur = s;
                acc = v;
            } else {
                acc.x += v.x; acc.y += v.y; acc.z += v.z; acc.w += v.w;
            }
        }
        e += 32;
    }
    if (cur >= 0)
        *(float4*)(my + (cur - s0) * DDIM + col) = acc;

    // ---- (b) GEMM: stripe(16x128) @ W(128x128), fp32 WMMA ----
    const int lo = lane & 15;
    const int hi = lane >> 4;             // 0 or 1
    v8f c[8] = {};
    const float* Arow = my + lo * DDIM;

    for (int k = 0; k < DDIM; k += 4) {
        v2f a = *(const v2f*)(Arow + k + 2 * hi);              // ds_load_b64
        // pair-row for this half-wave: (k+2*hi)/2 = k/2 + hi
        const float* w0 = &Wlds[((k >> 1) + hi) * WPITCH + lo * 2];
        #pragma unroll
        for (int t = 0; t < 8; ++t) {
            v2f b = *(const v2f*)(w0 + t * 32);                // ds_load_b64
            c[t] = __builtin_amdgcn_wmma_f32_16x16x4_f32(
                       false, a, false, b, (short)0, c[t], false, false);
        }
    }

    // ---- (c) tanh epilogue + store ----
    #pragma unroll
    for (int t = 0; t < 8; ++t) {
        #pragma unroll
        for (int r = 0; r < 8; ++r) {
            int m = s0 + r + 8 * hi;
            out[(size_t)m * DDIM + t * 16 + lo] = fast_tanh(c[t][r]);
        }
    }
}

// ---------------------------------------------------------------------------
extern "C" void kernel_launch(void* const* d_in, const int* in_sizes, int n_in,
                              void* d_out, int out_size, void* d_ws, size_t ws_size,
                              hipStream_t stream) {
    const float* values = (const float*)d_in[0];   // [N_SRC, 128] f32
    const float* W      = (const float*)d_in[1];   // [128, 128]  f32
    const int*   gidx   = (const int*)d_in[2];     // [E] int
    const int*   segid  = (const int*)d_in[3];     // [E] int, sorted
    float*       out    = (float*)d_out;           // [N_OUT, 128] f32
    int*         sstart = (int*)d_ws;              // [NSTRIPES] int scratch

    // 1) stripe_start init
    init_stripe_kernel<<<1, 256, 0, stream>>>(sstart);
    // 2) boundary marking sweep over sorted segment_ids
    stripe_start_kernel<<<(N_EDGES + 255) / 256, 256, 0, stream>>>(segid, sstart);
    // 3) fused gather+segsum (LDS) -> fp32 WMMA GEMM -> tanh
    fused_segsum_gemm_tanh_kernel<<<(NSTRIPES + WAVES_PB - 1) / WAVES_PB, 256, 0, stream>>>(
        values, gidx, segid, sstart, W, out);
}